// FCOSAssigner_70549132804501
// MI455X (gfx1250) — compile-verified
//
#include <hip/hip_runtime.h>
#include <hip/hip_bf16.h>
#include <stdint.h>

// FCOS target assigner for MI455X (gfx1250).
// Design: one thread per grid location (349,184 total). Each 256-thread block
// (8 wave32s) handles one batch's slice of locations; the 64 gt boxes (xywh,
// 1KB) + 64 labels for that batch are staged into LDS once per block with
// CDNA5 async global->LDS copies, synced with s_wait_asynccnt + barrier.
// All waves then broadcast-read the box table from LDS (conflict-free).

#define TOTAL_HW 21824   // 128^2 + 64^2 + 32^2 + 16^2 + 8^2
#define BIGF 99999999.0f

__global__ __launch_bounds__(256) void fcos_assign_kernel(
    const float* __restrict__ gt_boxes,  // [bs, 64, 4] xywh
    const int*   __restrict__ gt_labels, // [bs, 64]
    float*       __restrict__ out,       // cls[T] ++ cnt[T] ++ reg[T*4], T = bs*21824
    int bs)
{
    __shared__ float4 sBoxes[64];   // xywh
    __shared__ int    sLabels[64];

    const int blocksPerBatch = (TOTAL_HW + 255) / 256;   // 86
    const int b   = blockIdx.x / blocksPerBatch;
    const int loc = (blockIdx.x % blocksPerBatch) * 256 + threadIdx.x;
    const int tid = threadIdx.x;

    // ---- CDNA5 async stage: boxes (b128) + labels (b32) into LDS ----
    if (tid < 64) {
        uint32_t ldsB = (uint32_t)(uintptr_t)(&sBoxes[tid]);   // low 32 bits of flat addr == LDS offset
        uint32_t offB = (uint32_t)((b * 64 + tid) * 16);
        asm volatile("global_load_async_to_lds_b128 %0, %1, %2"
                     :: "v"(ldsB), "v"(offB), "s"(gt_boxes) : "memory");
        uint32_t ldsL = (uint32_t)(uintptr_t)(&sLabels[tid]);
        uint32_t offL = (uint32_t)((b * 64 + tid) * 4);
        asm volatile("global_load_async_to_lds_b32 %0, %1, %2"
                     :: "v"(ldsL), "v"(offL), "s"(gt_labels) : "memory");
    }
    asm volatile("s_wait_asynccnt 0" ::: "memory");  // gfx1250 split counter
    __syncthreads();

    if (loc >= TOTAL_HW) return;

    // ---- level lookup (branchless) ----
    const int   offs[5]  = {0, 16384, 20480, 21504, 21760};
    const int   lg2fw[5] = {7, 6, 5, 4, 3};
    const float los[5]   = {-1.f, 64.f, 128.f, 256.f, 512.f};
    const float his[5]   = {64.f, 128.f, 256.f, 512.f, 999999.f};
    const int lvl = (loc >= 16384) + (loc >= 20480) + (loc >= 21504) + (loc >= 21760);

    const int   hwin   = loc - offs[lvl];
    const int   lg     = lg2fw[lvl];
    const int   stride = 8 << lvl;
    const int   ix     = hwin & ((1 << lg) - 1);   // x varies fastest
    const int   iy     = hwin >> lg;
    const float s2     = 0.5f * (float)stride;
    const float x      = (float)ix * (float)stride + s2;
    const float y      = (float)iy * (float)stride + s2;
    const float lo     = los[lvl];
    const float hi     = his[lvl];
    const float radiu  = (float)stride * 1.5f;

    // ---- masked argmin over 64 boxes ----
    float best = BIGF;
    int   bi   = 0;
    float bl = 0.f, bt = 0.f, br = 0.f, bb = 0.f;
    bool  anyPos = false;

#pragma unroll 4
    for (int m = 0; m < 64; ++m) {
        const float4 bx = sBoxes[m];                 // ds_load_b128 broadcast
        const float x0 = bx.x, y0 = bx.y;
        const float x2 = bx.x + bx.z;                // xywh -> xyxy (same op order as ref)
        const float y2 = bx.y + bx.w;
        const float l = x  - x0;
        const float t = y  - y0;
        const float r = x2 - x;
        const float d = y2 - y;
        const float omin = fminf(fminf(l, t), fminf(r, d));
        const float omax = fmaxf(fmaxf(l, t), fmaxf(r, d));
        const float area = (l + r) * (t + d);
        const float cx = (x0 + x2) * 0.5f;
        const float cy = (y0 + y2) * 0.5f;
        const float coff = fmaxf(fabsf(x - cx), fabsf(y - cy));
        const bool pos = (omin > 0.f) & (omax > lo) & (omax <= hi) & (coff < radiu);
        anyPos |= pos;
        const float am = pos ? area : BIGF;
        if (am < best) { best = am; bi = m; bl = l; bt = t; br = r; bb = d; }  // first-min tie-break
    }

    // ---- write outputs ----
    const int TOT  = bs * TOTAL_HW;
    const int gidx = b * TOTAL_HW + loc;

    float  clsv = -1.f, cntv = -1.f;
    float4 regv = make_float4(-1.f, -1.f, -1.f, -1.f);
    if (anyPos) {
        clsv = (float)sLabels[bi];
        const float lrmin = fminf(bl, br), lrmax = fmaxf(bl, br);
        const float tbmin = fminf(bt, bb), tbmax = fmaxf(bt, bb);
        const float ratio = (lrmin * tbmin) / (lrmax * tbmax + 1e-10f);
        cntv = sqrtf(ratio);
        regv = make_float4(bl, bt, br, bb);
    }
    out[gidx]       = clsv;
    out[TOT + gidx] = cntv;
    float4* regOut = (float4*)(out + 2 * TOT);       // 16B-aligned: 2*TOT*4 bytes % 16 == 0
    regOut[gidx] = regv;
}

extern "C" void kernel_launch(void* const* d_in, const int* in_sizes, int n_in,
                              void* d_out, int out_size, void* d_ws, size_t ws_size,
                              hipStream_t stream) {
    const float* gt_boxes  = (const float*)d_in[0];   // [bs,64,4] float32 xywh
    const int*   gt_labels = (const int*)d_in[1];     // [bs,64] int32
    float* out = (float*)d_out;

    const int bs = in_sizes[1] / 64;                  // 16
    const int blocksPerBatch = (TOTAL_HW + 255) / 256; // 86
    dim3 grid(bs * blocksPerBatch);
    fcos_assign_kernel<<<grid, 256, 0, stream>>>(gt_boxes, gt_labels, out, bs);
}